// DecodePredictionsSoft_9947144258246
// MI455X (gfx1250) — compile-verified
//
#include <hip/hip_runtime.h>
#include <math.h>

// ---------------- problem constants (from reference) ----------------
#define BATCH        8
#define NUM_CLASSES  8
#define KMAX         100     // MAX_PER_CLASS
#define MAXDET       100
#define CONF_THR     0.05f
#define IOU_THR      0.5f
#define SIGMA        0.05f

#define P2_THREADS   1024    // 32 wave32 waves per workgroup
#define MAXJ         48      // ceil(49104 / 1024)
#define CAND         (NUM_CLASSES * KMAX)   // 800 per batch

#if defined(__has_builtin)
#  if __has_builtin(__builtin_amdgcn_tensor_load_to_lds) && \
      __has_builtin(__builtin_amdgcn_s_wait_tensorcnt)
#    define HAVE_TDM 1
#  else
#    define HAVE_TDM 0
#  endif
#else
#  define HAVE_TDM 0
#endif

#if HAVE_TDM
typedef unsigned int u32x4 __attribute__((ext_vector_type(4)));
typedef int          i32x4 __attribute__((ext_vector_type(4)));
typedef int          i32x8 __attribute__((ext_vector_type(8)));

// Low 32 bits of a flat LDS address are the LDS byte offset (ISA: the LDS
// aperture keeps the offset in addr[31:0]).
__device__ __forceinline__ unsigned lds_offset_of(const void* p) {
  return (unsigned)(unsigned long long)(uintptr_t)p;
}

// TDM descriptor per CDNA5 ISA.
// Gathers `rows` elements of 4B spaced `stride_elems` apart (a column of a
// row-major [rows, stride_elems] f32 tensor) into contiguous LDS at lds_off.
__device__ __forceinline__ void tdm_load_strided_column(
    unsigned lds_off, const float* gptr, int rows, int stride_elems) {
  unsigned long long ga = (unsigned long long)(uintptr_t)gptr;
  u32x4 g0;
  g0.x = 1u;                                             // count=1 (valid), user mode
  g0.y = lds_off;                                        // lds_addr [63:32]
  g0.z = (unsigned)(ga & 0xFFFFFFFFull);                 // global_addr [95:64]
  g0.w = (unsigned)((ga >> 32) & 0x01FFFFFFull)          // global_addr [120:96]
         | (2u << 30);                                   // type=2 ("image") [127:126]
  i32x8 g1;
  g1[0] = 0x00020000;                                    // data_size=2 -> 4 bytes
  g1[1] = (int)((unsigned)(stride_elems & 0xFFFF) << 16);// tensor_dim0[15:0] @ [63:48]
  g1[2] = (int)(((unsigned)(rows & 0xFFFF) << 16)        // tensor_dim1[15:0] @ [95:80]
         | ((unsigned)(stride_elems >> 16) & 0xFFFFu));  // tensor_dim0[31:16]
  g1[3] = (int)((1u << 16)                               // tile_dim0 = 1 @ [127:112]
         | (((unsigned)rows >> 16) & 0xFFFFu));          // tensor_dim1[31:16]
  g1[4] = (int)((unsigned)rows & 0xFFFFu);               // tile_dim1 = rows, tile_dim2=0
  g1[5] = stride_elems;                                  // tensor_dim0_stride[31:0]
  g1[6] = 0;
  g1[7] = 0;
  i32x4 z4 = {0, 0, 0, 0};
#if __clang_major__ >= 23
  i32x8 z8 = {0, 0, 0, 0, 0, 0, 0, 0};
  __builtin_amdgcn_tensor_load_to_lds(g0, g1, z4, z4, z8, 0);
#else
  __builtin_amdgcn_tensor_load_to_lds(g0, g1, z4, z4, 0);
#endif
}

// Contiguous 1-D global -> LDS DMA of `len` 4B elements (len < 65536).
__device__ __forceinline__ void tdm_load_contig(
    unsigned lds_off, const float* gptr, int len) {
  unsigned long long ga = (unsigned long long)(uintptr_t)gptr;
  u32x4 g0;
  g0.x = 1u;
  g0.y = lds_off;
  g0.z = (unsigned)(ga & 0xFFFFFFFFull);
  g0.w = (unsigned)((ga >> 32) & 0x01FFFFFFull) | (2u << 30);
  i32x8 g1;
  g1[0] = 0x00020000;                                    // data_size = 4 bytes
  g1[1] = (int)((unsigned)(len & 0xFFFF) << 16);         // tensor_dim0 = len
  g1[2] = (int)((1u << 16) | (((unsigned)len >> 16) & 0xFFFFu)); // tensor_dim1 = 1
  g1[3] = (int)((unsigned)(len & 0xFFFF) << 16);         // tile_dim0 = len
  g1[4] = 1;                                             // tile_dim1 = 1, tile_dim2 = 0
  g1[5] = len;                                           // tensor_dim0_stride = len
  g1[6] = 0;
  g1[7] = 0;
  i32x4 z4 = {0, 0, 0, 0};
#if __clang_major__ >= 23
  i32x8 z8 = {0, 0, 0, 0, 0, 0, 0, 0};
  __builtin_amdgcn_tensor_load_to_lds(g0, g1, z4, z4, z8, 0);
#else
  __builtin_amdgcn_tensor_load_to_lds(g0, g1, z4, z4, 0);
#endif
}
#endif // HAVE_TDM

// Monotone float <-> uint mapping so integer max == float max (handles -1.0).
__device__ __forceinline__ unsigned f2ord(float f) {
  unsigned u = __float_as_uint(f);
  return (u & 0x80000000u) ? ~u : (u | 0x80000000u);
}
__device__ __forceinline__ float ord2f(unsigned o) {
  unsigned u = (o & 0x80000000u) ? (o & 0x7fffffffu) : ~o;
  return __uint_as_float(u);
}
__device__ __forceinline__ unsigned long long pack_key(float s, int n) {
  return ((unsigned long long)f2ord(s) << 32) |
         (unsigned long long)(0xFFFFFFFFu - (unsigned)n);  // tie -> lowest index
}

// ---------------- phase 1: decode anchor boxes ----------------
__global__ void decode_boxes_kernel(const float* __restrict__ pred,
                                    const float* __restrict__ anchors,
                                    float4* __restrict__ boxesDec, int N) {
  int gid = blockIdx.x * blockDim.x + threadIdx.x;
  int total = BATCH * N;
  if (gid >= total) return;
  int n = gid % N;
  const float4 a = ((const float4*)anchors)[n];               // cx,cy,w,h
  const float4 p = *(const float4*)(pred + (size_t)gid * 12); // 16B aligned (12%4==0)
  float cx = p.x * a.z + a.x;
  float cy = p.y * a.w + a.y;
  float w  = expf(p.z) * a.z;
  float h  = expf(p.w) * a.w;
  boxesDec[gid] = make_float4(cx - 0.5f * w, cy - 0.5f * h,
                              cx + 0.5f * w, cy + 0.5f * h);
}

// ---------------- phase 2: soft-NMS, one workgroup per (batch, class) ----------------
// Record layout per (b,c,k): 8 floats {score, x0, y0, x1, y1, cls, valid, pad}
// Scores are owner-private per thread (thread t owns n = j*1024 + t), so the
// suppression pass also builds the NEXT step's local argmax key -> one LDS
// sweep and two barriers per step instead of two sweeps / three barriers.
__global__ __launch_bounds__(P2_THREADS)
void soft_nms_kernel(const float* __restrict__ pred,
                     const float4* __restrict__ boxesDec,
                     float* __restrict__ rec, int N) {
  // CDNA5: 320KB LDS per workgroup -> full 49104-float score vector on-chip
  __shared__ float s_scores[P2_THREADS * MAXJ];               // 196,608 B
  __shared__ unsigned long long s_part[P2_THREADS / 32];
  __shared__ unsigned long long s_best;

  const int t = threadIdx.x;
  const int b = blockIdx.x / NUM_CLASSES;
  const int c = blockIdx.x % NUM_CLASSES;
  const float4* bb = boxesDec + (size_t)b * N;
  const float*  pb = pred + (size_t)b * N * 12;

#if HAVE_TDM
  // TDM: DMA the strided class-c logit column [N x 1, stride 12] into LDS.
  if (t < 32) {
    tdm_load_strided_column(lds_offset_of(&s_scores[0]), pb + 4 + c, N, 12);
  }
#endif

  // pre-fill pad records (box 0, score 0, cls -1, valid 0) -- overlaps the DMA
  float* rbase = rec + (size_t)blockIdx.x * KMAX * 8;
  for (int k = t; k < KMAX; k += P2_THREADS) {
    float* r = rbase + k * 8;
    r[0] = 0.f; r[1] = 0.f; r[2] = 0.f; r[3] = 0.f;
    r[4] = 0.f; r[5] = -1.f; r[6] = 0.f; r[7] = 0.f;
  }

#if HAVE_TDM
  if (t < 32) __builtin_amdgcn_s_wait_tensorcnt(0);  // TENSORcnt is per-wave
  __syncthreads();                                   // publish DMA'd LDS
#endif

  // sigmoid (in place for the TDM path) + initial local argmax key
  unsigned long long key = 0ull;
  for (int j = 0; j < MAXJ; ++j) {
    int n = j * P2_THREADS + t;
    if (n < N) {
#if HAVE_TDM
      float x = s_scores[n];
#else
      float x = pb[(size_t)n * 12 + 4 + c];
#endif
      float s = 1.0f / (1.0f + expf(-x));
      s_scores[n] = s;
      unsigned long long kk = pack_key(s, n);
      if (kk > key) key = kk;
    }
  }

  const int lane = t & 31;
  const int wid  = t >> 5;

  for (int k = 0; k < KMAX; ++k) {
    // ---- workgroup argmax reduction of the thread-local keys ----
    unsigned long long wkey = key;
    #pragma unroll
    for (int off = 16; off > 0; off >>= 1) {
      unsigned long long o = __shfl_xor(wkey, off, 32);
      if (o > wkey) wkey = o;
    }
    if (lane == 0) s_part[wid] = wkey;
    __syncthreads();
    if (wid == 0) {
      unsigned long long kk = s_part[lane];
      #pragma unroll
      for (int off = 16; off > 0; off >>= 1) {
        unsigned long long o = __shfl_xor(kk, off, 32);
        if (o > kk) kk = o;
      }
      if (lane == 0) s_best = kk;
    }
    __syncthreads();

    unsigned long long best = s_best;
    float sc = ord2f((unsigned)(best >> 32));
    int   i  = (int)(0xFFFFFFFFu - (unsigned)(best & 0xFFFFFFFFull));
    if (!(sc > CONF_THR)) break;   // uniform across the workgroup

    float4 wb = bb[i];             // uniform-address load (L2 resident)

    if (t == 0) {
      // class id = argmax over the 8 logits of the selected anchor (first-wins)
      const float* lg = pb + (size_t)i * 12 + 4;
      int cb = 0; float vb = lg[0];
      #pragma unroll
      for (int cc = 1; cc < NUM_CLASSES; ++cc) {
        float v = lg[cc];
        if (v > vb) { vb = v; cb = cc; }
      }
      float* r = rbase + k * 8;
      r[0] = sc; r[1] = wb.x; r[2] = wb.y; r[3] = wb.z; r[4] = wb.w;
      r[5] = (float)cb; r[6] = 1.0f; r[7] = 0.f;
    }

    // ---- fused Gaussian suppression + next-step local argmax ----
    float areaW = (wb.z - wb.x) * (wb.w - wb.y);
    unsigned long long nkey = 0ull;
    for (int j = 0; j < MAXJ; ++j) {
      int n = j * P2_THREADS + t;
      if (n < N) {
        __builtin_prefetch(bb + n + P2_THREADS, 0, 0);   // global_prefetch_b8
        float4 bx = bb[n];
        float ltx = fmaxf(wb.x, bx.x), lty = fmaxf(wb.y, bx.y);
        float rbx = fminf(wb.z, bx.z), rby = fminf(wb.w, bx.w);
        float iw = fmaxf(rbx - ltx, 0.f), ih = fmaxf(rby - lty, 0.f);
        float inter = iw * ih;
        float areaB = (bx.z - bx.x) * (bx.w - bx.y);
        float iou = inter / fmaxf(areaW + areaB - inter, 1e-8f);
        float w = (iou <= IOU_THR) ? expf(iou * iou * (-0.5f / SIGMA)) : 0.f;
        float ns = (n == i) ? -1.0f : s_scores[n] * w;
        s_scores[n] = ns;
        unsigned long long kk = pack_key(ns, n);
        if (kk > nkey) nkey = kk;
      }
    }
    key = nkey;
    // no trailing barrier needed: s_scores is owner-private; s_part/s_best
    // accesses are fenced by the two reduction barriers above.
  }
}

// ---------------- phase 3: per-batch stable compaction / top-100 ----------------
__global__ void finalize_kernel(const float* __restrict__ rec,
                                float* __restrict__ out) {
  __shared__ float cd[CAND * 8];   // 25.6 KB
  const int t = threadIdx.x;
  const int b = blockIdx.x;
  const float* src = rec + (size_t)b * CAND * 8;

#if HAVE_TDM
  if (t < 32) {
    tdm_load_contig(lds_offset_of(&cd[0]), src, CAND * 8);  // 6400 x 4B contiguous
    __builtin_amdgcn_s_wait_tensorcnt(0);
  }
  __syncthreads();
#else
  const float4* src4 = (const float4*)src;
  float4* dst4 = (float4*)cd;
  for (int q = t; q < CAND * 2; q += blockDim.x) dst4[q] = src4[q];
  __syncthreads();
#endif

  int total = 0;
  for (int f = 0; f < CAND; ++f) total += (cd[f * 8 + 6] > 0.5f) ? 1 : 0;
  int vd = total < MAXDET ? total : MAXDET;
  if (t == 0) out[b] = (float)vd;                  // valid_detections

  const bool pathB = (total > MAXDET);
  float* oBoxes  = out + BATCH;
  float* oScores = out + BATCH + BATCH * MAXDET * 4;
  float* oCls    = oScores + BATCH * MAXDET;

  for (int f = t; f < CAND; f += blockDim.x) {
    if (!(cd[f * 8 + 6] > 0.5f)) continue;         // only valid entries placed
    float sc = cd[f * 8 + 0];
    int rank = 0;
    if (pathB) {
      // stable descending by score: earlier flat index wins ties
      for (int j = 0; j < CAND; ++j) {
        if (cd[j * 8 + 6] > 0.5f) {
          float sj = cd[j * 8 + 0];
          if (sj > sc || (sj == sc && j < f)) ++rank;
        }
      }
    } else {
      // stable compaction of valid entries in flat (class-major) order
      for (int j = 0; j < f; ++j) rank += (cd[j * 8 + 6] > 0.5f) ? 1 : 0;
    }
    if (rank < MAXDET) {
      float* rb = oBoxes + ((size_t)b * MAXDET + rank) * 4;
      rb[0] = cd[f * 8 + 1]; rb[1] = cd[f * 8 + 2];
      rb[2] = cd[f * 8 + 3]; rb[3] = cd[f * 8 + 4];
      oScores[b * MAXDET + rank] = sc;
      oCls[b * MAXDET + rank]    = cd[f * 8 + 5];
    }
  }

  // pad slots >= vd (boxes 0, scores 0, class -1)
  for (int k = t; k < MAXDET; k += blockDim.x) {
    if (k >= vd) {
      float* rb = oBoxes + ((size_t)b * MAXDET + k) * 4;
      rb[0] = 0.f; rb[1] = 0.f; rb[2] = 0.f; rb[3] = 0.f;
      oScores[b * MAXDET + k] = 0.f;
      oCls[b * MAXDET + k]    = -1.f;
    }
  }
}

// ---------------- launcher ----------------
extern "C" void kernel_launch(void* const* d_in, const int* in_sizes, int n_in,
                              void* d_out, int out_size, void* d_ws, size_t ws_size,
                              hipStream_t stream) {
  (void)n_in; (void)out_size; (void)ws_size;
  const float* pred    = (const float*)d_in[0];   // [8, N, 12]
  const float* anchors = (const float*)d_in[1];   // [N, 4]
  const int N = in_sizes[1] / 4;                  // 49104

  float* out = (float*)d_out;
  char*  ws  = (char*)d_ws;
  float4* boxesDec = (float4*)ws;                             // B*N float4
  size_t boxesBytes = (size_t)BATCH * N * sizeof(float4);     // ~6.3 MB
  float* rec = (float*)(ws + boxesBytes);                     // B*C*K*8 floats

  int total = BATCH * N;
  decode_boxes_kernel<<<(total + 255) / 256, 256, 0, stream>>>(
      pred, anchors, boxesDec, N);
  soft_nms_kernel<<<BATCH * NUM_CLASSES, P2_THREADS, 0, stream>>>(
      pred, boxesDec, rec, N);
  finalize_kernel<<<BATCH, 256, 0, stream>>>(rec, out);
}